// GCN_32925219291716
// MI455X (gfx1250) — compile-verified
//
#include <hip/hip_runtime.h>

#define N_NODES 10000
#define N_EDGES 640000
#define D 128          // D_IN == D_INTER == 128
#define NPAD 10240     // padded per-node arrays

typedef __attribute__((ext_vector_type(2))) float v2f;
typedef __attribute__((ext_vector_type(8))) float v8f;

// ---------------------------------------------------------------------------
// Step 1: zero per-node accumulators (deg reuses the dinv buffer)
// ---------------------------------------------------------------------------
__global__ void gcn_init(float* deg, int* counts, int* cursor) {
  int i = blockIdx.x * blockDim.x + threadIdx.x;
  if (i < N_NODES) { deg[i] = 0.0f; counts[i] = 0; cursor[i] = 0; }
}

// ---------------------------------------------------------------------------
// Step 2: weighted in-degree + in-edge counts (int/float atomics, once)
// edge_index layout: [2, E] row-major; row = ei[0..E), col = ei[E..2E)
// ---------------------------------------------------------------------------
__global__ void gcn_deg_count(const int* __restrict__ ei, const float* __restrict__ ew,
                              float* deg, int* counts) {
  int e = blockIdx.x * blockDim.x + threadIdx.x;
  if (e < N_EDGES) {
    int c = ei[N_EDGES + e];
    atomicAdd(&deg[c], ew[e]);
    atomicAdd(&counts[c], 1);
  }
}

// ---------------------------------------------------------------------------
// Step 3: dinv = rsqrt(deg + 1)   (+1 = self-loop weight; always > 0)
// ---------------------------------------------------------------------------
__global__ void gcn_dinv(float* dinv) {
  int i = blockIdx.x * blockDim.x + threadIdx.x;
  if (i < N_NODES) dinv[i] = rsqrtf(dinv[i] + 1.0f);
}

// ---------------------------------------------------------------------------
// Step 4: exclusive scan of counts -> CSR offsets (single workgroup, N=10000)
// ---------------------------------------------------------------------------
__global__ void gcn_scan(const int* __restrict__ counts, int* __restrict__ offsets) {
  __shared__ int ssum[256];
  const int chunk = (N_NODES + 255) / 256;  // 40
  const int tid = threadIdx.x;
  const int start = tid * chunk;
  int s = 0;
  for (int j = 0; j < chunk; ++j) {
    int idx = start + j;
    if (idx < N_NODES) s += counts[idx];
  }
  ssum[tid] = s;
  __syncthreads();
  if (tid == 0) {
    int acc = 0;
    for (int k = 0; k < 256; ++k) { int t = ssum[k]; ssum[k] = acc; acc += t; }
  }
  __syncthreads();
  int acc = ssum[tid];
  for (int j = 0; j < chunk; ++j) {
    int idx = start + j;
    if (idx < N_NODES) { offsets[idx] = acc; acc += counts[idx]; }
  }
  if (tid == 255) offsets[N_NODES] = acc;  // == N_EDGES
}

// ---------------------------------------------------------------------------
// Step 5: scatter edges into CSR (by target node) with precomputed norm
// ---------------------------------------------------------------------------
__global__ void gcn_fill_csr(const int* __restrict__ ei, const float* __restrict__ ew,
                             const float* __restrict__ dinv, const int* __restrict__ offsets,
                             int* cursor, int* __restrict__ csr_src, float* __restrict__ csr_w) {
  int e = blockIdx.x * blockDim.x + threadIdx.x;
  if (e < N_EDGES) {
    int r = ei[e];
    int c = ei[N_EDGES + e];
    float w = dinv[r] * ew[e] * dinv[c];
    int p = offsets[c] + atomicAdd(&cursor[c], 1);
    csr_src[p] = r;
    csr_w[p] = w;
  }
}

// ---------------------------------------------------------------------------
// Dense GEMM C[10000,128] = A[10000,128] @ B[128,128] in full fp32 using
// V_WMMA_F32_16X16X4_F32, M-blocked: each wave computes an 80x16 output slab
// (5 stacked 16x16 tiles sharing one B fragment per K-step).
//   - B loads (2x b32) amortized over 5 WMMAs -> ~1.4 VMEM per WMMA vs 3.
//   - 5 independent accumulators -> no WMMA->WMMA D->A/B RAW hazard, pipe
//     stays busy while the next K-step's loads are in flight.
// 10000 rows = 125 slabs x 80; x 8 N-tiles = 1000 full-EXEC waves.
// A 16x4 fp32 layout: lanes 0-15 hold M=lane,{K0,K1}; lanes 16-31 {K2,K3}.
// B 4x16 fp32 layout: symmetric. C/D: VGPR i -> M=i / M=i+8 per lane half.
// ---------------------------------------------------------------------------
__global__ void gcn_gemm_wmma(const float* __restrict__ A, const float* __restrict__ B,
                              float* __restrict__ C) {
  const int lane = threadIdx.x & 31;
  const int wave = threadIdx.x >> 5;
  const int wid  = blockIdx.x * 4 + wave;  // 0..999
  const int tm = wid >> 3;                 // 0..124 (80-row slab)
  const int tn = wid & 7;                  // 0..7
  const int m = lane & 15;
  const int half = lane >> 4;              // 0 or 1
  const int bcol = tn * 16 + m;

  const float* a0p = A + (size_t)(tm * 80 + m) * D;  // M-tile 0 row for this lane

  v8f c0 = {}, c1 = {}, c2 = {}, c3 = {}, c4 = {};
  for (int k = 0; k < D; k += 4) {
    const int ka = k + half * 2;
    v2f b;
    b.x = B[ka * D + bcol];
    b.y = B[(ka + 1) * D + bcol];
    v2f a0 = *(const v2f*)(a0p + ka);
    v2f a1 = *(const v2f*)(a0p + 16 * D + ka);
    v2f a2 = *(const v2f*)(a0p + 32 * D + ka);
    v2f a3 = *(const v2f*)(a0p + 48 * D + ka);
    v2f a4 = *(const v2f*)(a0p + 64 * D + ka);
    c0 = __builtin_amdgcn_wmma_f32_16x16x4_f32(false, a0, false, b, (short)0, c0, false, false);
    c1 = __builtin_amdgcn_wmma_f32_16x16x4_f32(false, a1, false, b, (short)0, c1, false, false);
    c2 = __builtin_amdgcn_wmma_f32_16x16x4_f32(false, a2, false, b, (short)0, c2, false, false);
    c3 = __builtin_amdgcn_wmma_f32_16x16x4_f32(false, a3, false, b, (short)0, c3, false, false);
    c4 = __builtin_amdgcn_wmma_f32_16x16x4_f32(false, a4, false, b, (short)0, c4, false, false);
  }

  float* crow = C + (size_t)(tm * 80 + half * 8) * D + bcol;
#pragma unroll
  for (int i = 0; i < 8; ++i) crow[i * D] = c0[i];
  crow += 16 * D;
#pragma unroll
  for (int i = 0; i < 8; ++i) crow[i * D] = c1[i];
  crow += 16 * D;
#pragma unroll
  for (int i = 0; i < 8; ++i) crow[i * D] = c2[i];
  crow += 16 * D;
#pragma unroll
  for (int i = 0; i < 8; ++i) crow[i * D] = c3[i];
  crow += 16 * D;
#pragma unroll
  for (int i = 0; i < 8; ++i) crow[i * D] = c4[i];
}

// ---------------------------------------------------------------------------
// CSR aggregation: out[v] = relu( dinv[v]^2 * h[v] + sum_in norm*h[src] + b )
// One wave per node; each lane owns 4 features (float4, 16B coalesced loads;
// a full wave covers the 512B row). global_prefetch issued PF edges ahead to
// break the dependent (index load -> row gather) chain for L2-resident rows.
// No float atomics anywhere.
// ---------------------------------------------------------------------------
#define PF 4
__global__ void gcn_aggregate(const float* __restrict__ h, const float* __restrict__ dinv,
                              const int* __restrict__ offsets, const int* __restrict__ csr_src,
                              const float* __restrict__ csr_w, const float* __restrict__ bias,
                              float* __restrict__ out) {
  const int lane = threadIdx.x & 31;
  const int v = (blockIdx.x * blockDim.x + threadIdx.x) >> 5;
  if (v >= N_NODES) return;
  const int f = lane * 4;

  const float dv = dinv[v];
  const float selfw = dv * dv;  // norm of self loop: dinv*1*dinv
  float4 acc = *(const float4*)(h + (size_t)v * D + f);
  acc.x *= selfw; acc.y *= selfw; acc.z *= selfw; acc.w *= selfw;

  const int beg = offsets[v];
  const int end = offsets[v + 1];
  for (int i = beg; i < end; ++i) {
    if (i + PF < end) {
      // pulls the (i+PF)'th source row's cacheline toward the WGP
      __builtin_prefetch(h + (size_t)csr_src[i + PF] * D + f, 0, 1);
    }
    const int s = csr_src[i];
    const float w = csr_w[i];
    const float4 hs = *(const float4*)(h + (size_t)s * D + f);
    acc.x = fmaf(w, hs.x, acc.x);
    acc.y = fmaf(w, hs.y, acc.y);
    acc.z = fmaf(w, hs.z, acc.z);
    acc.w = fmaf(w, hs.w, acc.w);
  }

  const float4 bb = *(const float4*)(bias + f);
  acc.x = fmaxf(acc.x + bb.x, 0.0f);
  acc.y = fmaxf(acc.y + bb.y, 0.0f);
  acc.z = fmaxf(acc.z + bb.z, 0.0f);
  acc.w = fmaxf(acc.w + bb.w, 0.0f);
  *(float4*)(out + (size_t)v * D + f) = acc;
}

// ---------------------------------------------------------------------------
extern "C" void kernel_launch(void* const* d_in, const int* in_sizes, int n_in,
                              void* d_out, int out_size, void* d_ws, size_t ws_size,
                              hipStream_t stream) {
  (void)in_sizes; (void)n_in; (void)out_size; (void)ws_size;

  const float* x  = (const float*)d_in[0];   // [10000,128]
  const int*   ei = (const int*)d_in[1];     // [2,640000] (int32 on device)
  const float* ew = (const float*)d_in[2];   // [640000]
  const float* W1 = (const float*)d_in[3];   // [128,128]
  const float* b1 = (const float*)d_in[4];   // [128]
  const float* W2 = (const float*)d_in[5];   // [128,128]
  const float* b2 = (const float*)d_in[6];   // [128]
  float* out = (float*)d_out;                // [10000,128]

  // workspace carve-up (~15.5 MB total)
  float* dinv    = (float*)d_ws;                      // NPAD floats (deg -> dinv)
  int*   counts  = (int*)(dinv + NPAD);               // NPAD
  int*   cursor  = counts + NPAD;                     // NPAD
  int*   offsets = cursor + NPAD;                     // NPAD (holds N+1)
  int*   csr_src = offsets + NPAD;                    // E
  float* csr_w   = (float*)(csr_src + N_EDGES);       // E
  float* hbuf    = csr_w + N_EDGES;                   // N*D
  float* gbuf    = hbuf + (size_t)N_NODES * D;        // N*D

  // --- build normalization + CSR once (shared by both layers) ---
  gcn_init<<<(N_NODES + 255) / 256, 256, 0, stream>>>(dinv, counts, cursor);
  gcn_deg_count<<<(N_EDGES + 255) / 256, 256, 0, stream>>>(ei, ew, dinv, counts);
  gcn_dinv<<<(N_NODES + 255) / 256, 256, 0, stream>>>(dinv);
  gcn_scan<<<1, 256, 0, stream>>>(counts, offsets);
  gcn_fill_csr<<<(N_EDGES + 255) / 256, 256, 0, stream>>>(ei, ew, dinv, offsets,
                                                          cursor, csr_src, csr_w);

  // --- layer 1: H = X@W1 ; g = relu(aggregate(H) + b1) ---
  gcn_gemm_wmma<<<250, 128, 0, stream>>>(x, W1, hbuf);
  gcn_aggregate<<<2500, 128, 0, stream>>>(hbuf, dinv, offsets, csr_src, csr_w, b1, gbuf);

  // --- layer 2: H = g@W2 ; out = relu(aggregate(H) + b2) ---
  gcn_gemm_wmma<<<250, 128, 0, stream>>>(gbuf, W2, hbuf);
  gcn_aggregate<<<2500, 128, 0, stream>>>(hbuf, dinv, offsets, csr_src, csr_w, b2, out);
}